// GPT2PagedAttention_86655260165006
// MI455X (gfx1250) — compile-verified
//
#include <hip/hip_runtime.h>

typedef __attribute__((ext_vector_type(2))) float v2f;
typedef __attribute__((ext_vector_type(4))) float v4f;
typedef __attribute__((ext_vector_type(8))) float v8f;
typedef __attribute__((ext_vector_type(4))) int   v4i;

#define B_   32
#define D_   1024
#define H_   16
#define DH_  64
#define KV_  8192
#define TPB  256
#define KTILE   128            // K rows staged per async tile
#define KPITCH  68             // padded row pitch (floats): 272B, 16B aligned

// async-copy builtin wants generic int4* for both global src and LDS dst
#define GP128(p) ((v4i*)(p))

// ---------------------------------------------------------------------------
// WMMA fp32 GEMM: C[M,N] = A[M,K] @ B[K,N] + bias[N], M = 32 (two 16-row tiles)
// One wave per 16x16 tile; V_WMMA_F32_16X16X4_F32 K-loop, pointer-incremented.
// ---------------------------------------------------------------------------
__global__ __launch_bounds__(32) void wmma_gemm_bias_kernel(
    const float* __restrict__ A, const float* __restrict__ Bm,
    const float* __restrict__ bias, float* __restrict__ C,
    int N, int K) {
  const int ntile = blockIdx.x;
  const int mtile = blockIdx.y;
  const int lane  = threadIdx.x & 31;
  const int half  = lane >> 4;   // 0: lanes 0-15, 1: lanes 16-31
  const int idx   = lane & 15;

  const float* __restrict__ aP =
      A + (size_t)(mtile * 16 + idx) * K + 2 * half;
  const float* __restrict__ bP =
      Bm + (ntile * 16 + idx) + (size_t)(2 * half) * N;

  v8f c = {};
#pragma unroll 4
  for (int k = 0; k < K; k += 4) {
    v2f a, b;
    a.x = aP[0];
    a.y = aP[1];
    b.x = bP[0];
    b.y = bP[N];
    c = __builtin_amdgcn_wmma_f32_16x16x4_f32(
        false, a, false, b, (short)0, c, false, false);
    aP += 4;
    bP += (size_t)4 * N;
  }

  const float bv = bias[ntile * 16 + idx];
  float* __restrict__ Cout =
      C + (size_t)(mtile * 16 + 8 * half) * N + ntile * 16 + idx;
#pragma unroll
  for (int j = 0; j < 8; ++j) Cout[(size_t)j * N] = c[j] + bv;
}

// ---------------------------------------------------------------------------
// Decode attention, one workgroup per (b,h).
// Pass 1: producer waves (tid 128..255) stream K tiles into LDS with
//         GLOBAL_LOAD_ASYNC_TO_LDS_B128 (coalesced, double-buffered,
//         ASYNCcnt-synchronized); consumer waves (tid 0..127) compute one
//         q.K[row] dot each per tile out of LDS.
// Pass 2: coalesced non-temporal V stream, threads = (d, row-group).
// ---------------------------------------------------------------------------
__global__ __launch_bounds__(TPB) void attn_kernel(
    const float* __restrict__ k_cache, const float* __restrict__ v_cache,
    const float* __restrict__ qkv, float* __restrict__ context) {
  __shared__ float s_e[KV_ + 8];                      // scores -> exp weights
  __shared__ __align__(16) float kbuf[2][KTILE * KPITCH];
  __shared__ __align__(16) float s_q[DH_];
  __shared__ float s_red[TPB];
  __shared__ float s_acc[4][DH_];

  const int bh  = blockIdx.x;      // 0..511
  const int b   = bh >> 4;
  const int h   = bh & 15;
  const int tid = threadIdx.x;

  const float* __restrict__ qptr = qkv + (size_t)b * (3 * D_) + h * DH_;
  const float* __restrict__ kcur = qptr + D_;       // k_curr slice of qkv
  const float* __restrict__ vcur = qptr + 2 * D_;   // v_curr slice of qkv
  const float* __restrict__ Kb   = k_cache + (size_t)bh * KV_ * DH_;
  const float* __restrict__ Vb   = v_cache + (size_t)bh * KV_ * DH_;

  if (tid < DH_) s_q[tid] = qptr[tid];
  __syncthreads();

  v4f qv[16];
#pragma unroll
  for (int i = 0; i < 16; ++i) qv[i] = ((const v4f*)s_q)[i];

  const bool is_prod = (tid >= 128);
  const int  p       = tid & 127;
  const int  n_tiles = KV_ / KTILE;   // 64

  // ---- Pass 1: async-staged K stream + per-row dot products ----
  if (is_prod) {
    // issue tile 0 into buffer 0: 128x64 f32 = 2048 b128 chunks, 16/thread
    const float* gsrc = Kb;
#pragma unroll
    for (int it = 0; it < 16; ++it) {
      const int j = it * 128 + p;             // chunk id, lane-consecutive
      const int row = j >> 4, cc = j & 15;
      __builtin_amdgcn_global_load_async_to_lds_b128(
          GP128(gsrc + (size_t)j * 4),
          GP128(&kbuf[0][row * KPITCH + cc * 4]), 0, 0);
    }
  }

  float lmax = -__builtin_inff();
  for (int t = 0; t < n_tiles; ++t) {
    if (is_prod) __builtin_amdgcn_s_wait_asynccnt(0);  // tile t landed in LDS
    __syncthreads();                                   // publish buffer t&1
    const int bs = t & 1;
    if (is_prod) {
      if (t + 1 < n_tiles) {                           // prefetch tile t+1
        const float* gsrc = Kb + (size_t)(t + 1) * KTILE * DH_;
#pragma unroll
        for (int it = 0; it < 16; ++it) {
          const int j = it * 128 + p;
          const int row = j >> 4, cc = j & 15;
          __builtin_amdgcn_global_load_async_to_lds_b128(
              GP128(gsrc + (size_t)j * 4),
              GP128(&kbuf[bs ^ 1][row * KPITCH + cc * 4]), 0, 0);
        }
      }
    } else {
      // consumer: dot(q, K[t*KTILE + tid]) from LDS
      const v4f* __restrict__ kr = (const v4f*)&kbuf[bs][tid * KPITCH];
      v4f a4 = {};
#pragma unroll
      for (int i = 0; i < 16; ++i) a4 += qv[i] * kr[i];
      const float s = a4.x + a4.y + a4.z + a4.w;
      s_e[t * KTILE + tid] = s;
      lmax = fmaxf(lmax, s);
    }
  }
  if (tid == 0) {                                      // appended row (k_curr)
    const v4f* __restrict__ kr = (const v4f*)kcur;
    v4f a4 = {};
#pragma unroll
    for (int i = 0; i < 16; ++i) a4 += qv[i] * kr[i];
    const float s = a4.x + a4.y + a4.z + a4.w;
    s_e[KV_] = s;
    lmax = fmaxf(lmax, s);
  }

  // ---- max reduction ----
  s_red[tid] = lmax;
  __syncthreads();
  for (int off = TPB / 2; off > 0; off >>= 1) {
    if (tid < off) s_red[tid] = fmaxf(s_red[tid], s_red[tid + off]);
    __syncthreads();
  }
  const float m = s_red[0];
  __syncthreads();

  // ---- exp + sum reduction ----
  float lsum = 0.f;
  for (int kk = tid; kk < KV_ + 1; kk += TPB) {
    const float e = __expf(s_e[kk] - m);
    s_e[kk] = e;
    lsum += e;
  }
  s_red[tid] = lsum;
  __syncthreads();
  for (int off = TPB / 2; off > 0; off >>= 1) {
    if (tid < off) s_red[tid] += s_red[tid + off];
    __syncthreads();
  }
  const float l = s_red[0];

  // ---- Pass 2: acc[d] = sum_k e[k] * V[k][d]; threads = (d, group of 4) ----
  const int d = tid & (DH_ - 1);
  const int g = tid >> 6;
  float acc = 0.f;
  for (int kk = g; kk < KV_; kk += 4) {
    if (kk + 32 < KV_)
      __builtin_prefetch(Vb + (size_t)(kk + 32) * DH_ + d, 0, 0);
    acc += s_e[kk] * __builtin_nontemporal_load(Vb + (size_t)kk * DH_ + d);
  }
  if (g == (KV_ & 3)) acc += s_e[KV_] * vcur[d];       // appended row (v_curr)
  s_acc[g][d] = acc;
  __syncthreads();

  if (tid < DH_) {
    const float ctx = (s_acc[0][tid] + s_acc[1][tid] +
                       s_acc[2][tid] + s_acc[3][tid]) / fmaxf(l, 1e-9f);
    context[(size_t)b * D_ + h * DH_ + tid] = ctx;
  }
}

// ---------------------------------------------------------------------------
extern "C" void kernel_launch(void* const* d_in, const int* in_sizes, int n_in,
                              void* d_out, int out_size, void* d_ws, size_t ws_size,
                              hipStream_t stream) {
  const float* hidden  = (const float*)d_in[0];
  const float* k_cache = (const float*)d_in[1];
  const float* v_cache = (const float*)d_in[2];
  const float* w_attn  = (const float*)d_in[3];
  const float* b_attn  = (const float*)d_in[4];
  const float* w_proj  = (const float*)d_in[5];
  const float* b_proj  = (const float*)d_in[6];
  float* out = (float*)d_out;

  float* qkv     = (float*)d_ws;                    // [32, 3072]
  float* context = qkv + (size_t)B_ * 3 * D_;       // [32, 1024]

  // 1) qkv = hidden @ w_attn + b_attn   (WMMA f32, tiles: 192 x 2)
  dim3 g1(3 * D_ / 16, B_ / 16);
  wmma_gemm_bias_kernel<<<g1, 32, 0, stream>>>(hidden, w_attn, b_attn, qkv,
                                               3 * D_, D_);

  // 2) streaming decode attention, one workgroup per (b,h)
  attn_kernel<<<B_ * H_, TPB, 0, stream>>>(k_cache, v_cache, qkv, context);

  // 3) out = context @ w_proj + b_proj  (WMMA f32, tiles: 64 x 2)
  dim3 g3(D_ / 16, B_ / 16);
  wmma_gemm_bias_kernel<<<g3, 32, 0, stream>>>(context, w_proj, b_proj, out,
                                               D_, D_);
}